// HeteroGNN_63247688401689
// MI455X (gfx1250) — compile-verified
//
#include <hip/hip_runtime.h>
#include <math.h>

typedef float v2f __attribute__((ext_vector_type(2)));
typedef float v8f __attribute__((ext_vector_type(8)));
typedef unsigned int u32x4 __attribute__((ext_vector_type(4)));
typedef int i32x4 __attribute__((ext_vector_type(4)));
typedef int i32x8 __attribute__((ext_vector_type(8)));

#define D_FEAT 128
#define NREL 8
#define TILE_E 64
#define A_STRIDE 132   // (m*132)%64 == 4m  -> conflict-free 16x4 A-fragment reads
#define W_STRIDE 136   // (k*136)%64 == 8k; k vs k+2 rows land 16 banks apart -> conflict-free B reads

// ---------------------------------------------------------------------------
// TDM: issue a 2D tensor_load_to_lds with LDS row padding.
// D# layout per CDNA5 ISA 8.3/8.4 (group0 128b, group1 256b; groups 2/3 zero).
// pad_interval code: 0=2,1=4,...,6=128 DWORDs; pad_amount code: v+1 DWORDs.
// This toolchain exposes the 6-arg builtin: (g0, g1, g2, g3, g4, cpol).
// ---------------------------------------------------------------------------
__device__ __forceinline__ void tdm_load_2d(const void* gaddr, unsigned lds_off,
                                            unsigned td0, unsigned td1,
                                            unsigned tile0, unsigned tile1,
                                            unsigned stride0,
                                            unsigned pad_interval,
                                            unsigned pad_amount) {
  unsigned long long ga = (unsigned long long)(uintptr_t)gaddr;
  u32x4 g0;
  g0.x = 1u;                                                  // count=1 (valid), no gather
  g0.y = lds_off;                                             // lds_addr (bytes)
  g0.z = (unsigned)(ga & 0xFFFFFFFFu);                        // global_addr[31:0]
  g0.w = (unsigned)((ga >> 32) & 0x01FFFFFFu) | (2u << 30);   // global_addr[56:32], type=2
  i32x8 g1;
  g1[0] = (int)((2u << 16) |                                  // data_size = 4B
                (1u << 20) |                                  // pad_enable
                (pad_interval << 22) | (pad_amount << 25));
  g1[1] = (int)((td0 & 0xFFFFu) << 16);                       // tensor_dim0[15:0] @bit48
  g1[2] = (int)((td0 >> 16) | ((td1 & 0xFFFFu) << 16));       // td0[31:16], td1[15:0]
  g1[3] = (int)((td1 >> 16) | (tile0 << 16));                 // td1[31:16], tile_dim0
  g1[4] = (int)tile1;                                         // tile_dim1; tile_dim2=0
  g1[5] = (int)stride0;                                       // tensor_dim0_stride[31:0]
  g1[6] = 0;                                                  // stride0[47:32], stride1[15:0]
  g1[7] = 0;
  i32x4 gz4 = {0, 0, 0, 0};
  i32x8 gz8 = {0, 0, 0, 0, 0, 0, 0, 0};
  __builtin_amdgcn_tensor_load_to_lds(g0, g1, gz4, gz4, gz8, 0);
}

// ---------------------------------------------------------------------------
// Pass 1: per-(relation,dst) edge counts + per-relation histogram
// ---------------------------------------------------------------------------
__global__ __launch_bounds__(256) void k_count(const int* __restrict__ etype,
                                               const int* __restrict__ edst,
                                               unsigned* __restrict__ counts,
                                               unsigned* __restrict__ relCount,
                                               int E, int N) {
  int e = blockIdx.x * blockDim.x + threadIdx.x;
  if (e >= E) return;
  int r = etype[e];
  int d = edst[e];
  atomicAdd(&counts[(size_t)r * N + d], 1u);
  atomicAdd(&relCount[r], 1u);
}

// ---------------------------------------------------------------------------
// Pass 2: 8-entry exclusive prefix sums (edge ranges + padded tile table)
// ---------------------------------------------------------------------------
__global__ void k_prefix(const unsigned* __restrict__ relCount,
                         unsigned* __restrict__ relStart,
                         unsigned* __restrict__ tileBase) {
  if (threadIdx.x == 0) {
    unsigned s = 0, t = 0;
    for (int r = 0; r < NREL; ++r) {
      relStart[r] = s;
      tileBase[r] = t;
      s += relCount[r];
      t += (relCount[r] + TILE_E - 1) / TILE_E;
    }
    tileBase[NREL] = t;
  }
}

// ---------------------------------------------------------------------------
// Pass 3: scatter edge ids into relation-sorted order
// ---------------------------------------------------------------------------
__global__ __launch_bounds__(256) void k_scatter(const int* __restrict__ etype,
                                                 const unsigned* __restrict__ relStart,
                                                 unsigned* __restrict__ relCursor,
                                                 int* __restrict__ sorted, int E) {
  int e = blockIdx.x * blockDim.x + threadIdx.x;
  if (e >= E) return;
  int r = etype[e];
  unsigned pos = relStart[r] + atomicAdd(&relCursor[r], 1u);
  sorted[pos] = e;
}

// ---------------------------------------------------------------------------
// Pass 4: WMMA edge GEMM: msg = (x[src] @ W[r]) / cnt, atomically summed into
// agg (= d_out). W[r] staged to LDS by the TDM; A rows gathered manually.
// Block = 4 waves; each wave: 16 edges x 128 cols.
// ---------------------------------------------------------------------------
__global__ __launch_bounds__(128) void k_edge_gemm(
    const float* __restrict__ x, const float* __restrict__ W,
    const int* __restrict__ esrc, const int* __restrict__ edst,
    const unsigned* __restrict__ counts, const int* __restrict__ sorted,
    const unsigned* __restrict__ relCount, const unsigned* __restrict__ relStart,
    const unsigned* __restrict__ tileBase,
    float* __restrict__ out, int N) {
  __shared__ float sW[D_FEAT * W_STRIDE];   // 69,632 B
  __shared__ float sA[TILE_E * A_STRIDE];   // 33,792 B
  __shared__ int sSrc[TILE_E];
  __shared__ int sDst[TILE_E];
  __shared__ float sInv[TILE_E];

  unsigned tile = blockIdx.x;
  if (tile >= tileBase[NREL]) return;  // uniform per block

  int r = 0;
#pragma unroll
  for (int i = 1; i < NREL; ++i) r = (tile >= tileBase[i]) ? i : r;
  unsigned lt = tile - tileBase[r];
  unsigned base = relStart[r] + lt * TILE_E;
  int nvalid = (int)relCount[r] - (int)(lt * TILE_E);
  if (nvalid > TILE_E) nvalid = TILE_E;

  int tid = threadIdx.x;

  // wave 0 kicks the TDM for W[r]: 128x128 f32 tile, LDS rows padded to 136 f
  if (tid < 32) {
    tdm_load_2d(W + (size_t)r * D_FEAT * D_FEAT,
                (unsigned)(uintptr_t)sW,
                D_FEAT, D_FEAT, D_FEAT, D_FEAT, D_FEAT,
                /*interval 128 dw*/ 6u, /*amount 8 dw*/ 7u);
  }

  if (tid < TILE_E) {
    int s = 0, d = 0;
    float inv = 0.0f;
    if (tid < nvalid) {
      int e = sorted[base + tid];
      s = esrc[e];
      d = edst[e];
      inv = 1.0f / (float)counts[(size_t)r * N + d];  // >=1 for real edges
    }
    sSrc[tid] = s; sDst[tid] = d; sInv[tid] = inv;
  }
  __syncthreads();

  // coalesced gather of 64 source rows into LDS
  for (int idx = tid; idx < TILE_E * D_FEAT; idx += 128) {
    int row = idx >> 7, col = idx & 127;
    sA[row * A_STRIDE + col] = x[(size_t)sSrc[row] * D_FEAT + col];
  }
  if (tid < 32) __builtin_amdgcn_s_wait_tensorcnt(0);
  __syncthreads();

  int lane = tid & 31, w = tid >> 5;
  int m = (w << 4) + (lane & 15);
  int kb = (lane >> 4) << 1;  // 0 or 2
  int n = lane & 15;

  v8f z = {0.f, 0.f, 0.f, 0.f, 0.f, 0.f, 0.f, 0.f};
  v8f acc[8];
#pragma unroll
  for (int i = 0; i < 8; ++i) acc[i] = z;

#pragma unroll 4
  for (int k0 = 0; k0 < D_FEAT; k0 += 4) {
    v2f a = *(const v2f*)&sA[m * A_STRIDE + k0 + kb];
    const float* wp = &sW[(k0 + kb) * W_STRIDE + n];
#pragma unroll
    for (int nt = 0; nt < 8; ++nt) {
      v2f b;
      b.x = wp[nt * 16];
      b.y = wp[W_STRIDE + nt * 16];
      acc[nt] = __builtin_amdgcn_wmma_f32_16x16x4_f32(
          false, a, false, b, (short)0, acc[nt], false, false);
    }
  }

  int hi = (lane >> 4) << 3;  // 0 or 8
#pragma unroll
  for (int nt = 0; nt < 8; ++nt) {
#pragma unroll
    for (int v = 0; v < 8; ++v) {
      int rr = (w << 4) + v + hi;
      if (rr < nvalid) {
        atomicAdd(&out[(size_t)sDst[rr] * D_FEAT + nt * 16 + n],
                  acc[nt][v] * sInv[rr]);
      }
    }
  }
}

// ---------------------------------------------------------------------------
// Pass 5: WMMA root GEMM fused with +agg +bias (out = x@root + out + bias).
// Both A (contiguous rows) and root staged by the TDM.
// ---------------------------------------------------------------------------
__global__ __launch_bounds__(128) void k_root_gemm(
    const float* __restrict__ x, const float* __restrict__ root,
    const float* __restrict__ bias, float* __restrict__ out, int N) {
  __shared__ float sW[D_FEAT * W_STRIDE];
  __shared__ float sA[TILE_E * A_STRIDE];
  int row0 = blockIdx.x * TILE_E;
  int tid = threadIdx.x;

  if (tid < 32) {
    tdm_load_2d(root, (unsigned)(uintptr_t)sW,
                D_FEAT, D_FEAT, D_FEAT, D_FEAT, D_FEAT, 6u, 7u);
    // tensor_dim1 = N-row0: tail rows beyond the tensor read back as zero
    tdm_load_2d(x + (size_t)row0 * D_FEAT, (unsigned)(uintptr_t)sA,
                D_FEAT, (unsigned)(N - row0), D_FEAT, TILE_E, D_FEAT,
                /*interval 128 dw*/ 6u, /*amount 4 dw*/ 3u);
    __builtin_amdgcn_s_wait_tensorcnt(0);
  }
  __syncthreads();

  int lane = tid & 31, w = tid >> 5;
  int m = (w << 4) + (lane & 15);
  int kb = (lane >> 4) << 1;
  int n = lane & 15;

  v8f z = {0.f, 0.f, 0.f, 0.f, 0.f, 0.f, 0.f, 0.f};
  v8f acc[8];
#pragma unroll
  for (int i = 0; i < 8; ++i) acc[i] = z;

#pragma unroll 4
  for (int k0 = 0; k0 < D_FEAT; k0 += 4) {
    v2f a = *(const v2f*)&sA[m * A_STRIDE + k0 + kb];
    const float* wp = &sW[(k0 + kb) * W_STRIDE + n];
#pragma unroll
    for (int nt = 0; nt < 8; ++nt) {
      v2f b;
      b.x = wp[nt * 16];
      b.y = wp[W_STRIDE + nt * 16];
      acc[nt] = __builtin_amdgcn_wmma_f32_16x16x4_f32(
          false, a, false, b, (short)0, acc[nt], false, false);
    }
  }

  int hi = (lane >> 4) << 3;
#pragma unroll
  for (int nt = 0; nt < 8; ++nt) {
#pragma unroll
    for (int v = 0; v < 8; ++v) {
      int gr = row0 + (w << 4) + v + hi;
      if (gr < N) {
        int col = nt * 16 + n;
        size_t o = (size_t)gr * D_FEAT + col;
        out[o] = acc[nt][v] + out[o] + bias[col];
      }
    }
  }
}

// ---------------------------------------------------------------------------
// Pass 6: per-column sum / sum-of-squares (stride is multiple of 128)
// ---------------------------------------------------------------------------
__global__ __launch_bounds__(256) void k_bn_stats(const float* __restrict__ out,
                                                  float* __restrict__ colSum,
                                                  float* __restrict__ colSq,
                                                  long long total) {
  long long stride = (long long)gridDim.x * blockDim.x;
  long long i0 = (long long)blockIdx.x * blockDim.x + threadIdx.x;
  int col = (int)(i0 & (D_FEAT - 1));
  float s = 0.f, q = 0.f;
  for (long long i = i0; i < total; i += stride) {
    float v = out[i];
    s += v;
    q += v * v;
  }
  atomicAdd(&colSum[col], s);
  atomicAdd(&colSq[col], q);
}

// ---------------------------------------------------------------------------
// Pass 7: BatchNorm (biased var) + Mish, in place on d_out
// ---------------------------------------------------------------------------
__global__ __launch_bounds__(256) void k_bn_mish(float* __restrict__ out,
                                                 const float* __restrict__ colSum,
                                                 const float* __restrict__ colSq,
                                                 const float* __restrict__ gamma,
                                                 const float* __restrict__ beta,
                                                 int N) {
  long long i = (long long)blockIdx.x * blockDim.x + threadIdx.x;
  long long total = (long long)N * D_FEAT;
  if (i >= total) return;
  int col = (int)(i & (D_FEAT - 1));
  float invN = 1.0f / (float)N;
  float mean = colSum[col] * invN;
  float var = colSq[col] * invN - mean * mean;
  float v = out[i];
  float xn = (v - mean) * rsqrtf(var + 1e-5f) * gamma[col] + beta[col];
  float sp = (xn > 20.f) ? xn : log1pf(__expf(xn));
  out[i] = xn * tanhf(sp);
}

// ---------------------------------------------------------------------------
extern "C" void kernel_launch(void* const* d_in, const int* in_sizes, int n_in,
                              void* d_out, int out_size, void* d_ws, size_t ws_size,
                              hipStream_t stream) {
  const float* x = (const float*)d_in[0];
  const int* ei = (const int*)d_in[1];
  const int* et = (const int*)d_in[2];
  const float* W = (const float*)d_in[3];
  const float* root = (const float*)d_in[4];
  const float* bias = (const float*)d_in[5];
  const float* gamma = (const float*)d_in[6];
  const float* beta = (const float*)d_in[7];
  float* out = (float*)d_out;

  int N = in_sizes[0] / D_FEAT;
  int E = in_sizes[2];
  const int* esrc = ei;
  const int* edst = ei + E;

  // workspace layout: [counts 8N][relCount 8][relCursor 8][relStart 8]
  //                   [tileBase 16][colSum 128][colSq 128][sorted E]
  char* wsb = (char*)d_ws;
  size_t off = 0;
  unsigned* counts = (unsigned*)(wsb + off);    off += (size_t)NREL * N * 4;
  unsigned* relCount = (unsigned*)(wsb + off);  off += 8 * 4;
  unsigned* relCursor = (unsigned*)(wsb + off); off += 8 * 4;
  unsigned* relStart = (unsigned*)(wsb + off);  off += 8 * 4;
  unsigned* tileBase = (unsigned*)(wsb + off);  off += 16 * 4;
  float* colSum = (float*)(wsb + off);          off += D_FEAT * 4;
  float* colSq = (float*)(wsb + off);           off += D_FEAT * 4;
  size_t zeroBytes = off;
  int* sorted = (int*)(wsb + off);              off += (size_t)E * 4;
  (void)n_in; (void)ws_size;

  (void)hipMemsetAsync(d_ws, 0, zeroBytes, stream);
  (void)hipMemsetAsync(d_out, 0, (size_t)out_size * sizeof(float), stream);

  int eblocks = (E + 255) / 256;
  k_count<<<eblocks, 256, 0, stream>>>(et, edst, counts, relCount, E, N);
  k_prefix<<<1, 32, 0, stream>>>(relCount, relStart, tileBase);
  k_scatter<<<eblocks, 256, 0, stream>>>(et, relStart, relCursor, sorted, E);

  int maxTiles = (E + TILE_E - 1) / TILE_E + NREL;  // padded upper bound
  k_edge_gemm<<<maxTiles, 128, 0, stream>>>(x, W, esrc, edst, counts, sorted,
                                            relCount, relStart, tileBase, out, N);

  int rblocks = (N + TILE_E - 1) / TILE_E;
  k_root_gemm<<<rblocks, 128, 0, stream>>>(x, root, bias, out, N);

  k_bn_stats<<<512, 256, 0, stream>>>(out, colSum, colSq, (long long)N * D_FEAT);

  long long total = (long long)N * D_FEAT;
  int mblocks = (int)((total + 255) / 256);
  k_bn_mish<<<mblocks, 256, 0, stream>>>(out, colSum, colSq, gamma, beta, N);
}